// SAGEConv_56908316672596
// MI455X (gfx1250) — compile-verified
//
#include <hip/hip_runtime.h>
#include <hip/hip_bf16.h>

typedef float v2f __attribute__((ext_vector_type(2)));
typedef float v8f __attribute__((ext_vector_type(8)));

#define DIM 128   // feature dim (in == out == 128)

// ---------------------------------------------------------------------------
// Kernel 1: zero the scratch accumulators (summed[N*128] ++ countf[N])
// ---------------------------------------------------------------------------
__global__ void sage_zero_kernel(float* __restrict__ p, size_t n) {
    size_t i = (size_t)blockIdx.x * blockDim.x + threadIdx.x;
    if (i < n) p[i] = 0.0f;
}

// ---------------------------------------------------------------------------
// Kernel 2: edge scatter-add. One wave32 per edge: 32 lanes x float4 = 128
// floats. Gather x[src] (coalesced 512B) -> atomicAdd into summed[dst].
// Accumulator fits in the 192MB L2, so atomics stay on-chip.
// ---------------------------------------------------------------------------
__global__ void sage_scatter_kernel(const float* __restrict__ x,
                                    const long long* __restrict__ ei,
                                    float* __restrict__ summed,
                                    float* __restrict__ countf,
                                    int nEdges) {
    int edge = (int)(((size_t)blockIdx.x * blockDim.x + threadIdx.x) >> 5);
    int lane = threadIdx.x & 31;
    if (edge >= nEdges) return;

    long long src = ei[edge];                       // edge_index[0][e]
    long long dst = ei[(size_t)nEdges + edge];      // edge_index[1][e]

    const float4* xr = (const float4*)(x + (size_t)src * DIM);
    float4 v = xr[lane];                            // lanes cover 128 floats

    float* srow = summed + (size_t)dst * DIM + lane * 4;
    unsafeAtomicAdd(srow + 0, v.x);
    unsafeAtomicAdd(srow + 1, v.y);
    unsafeAtomicAdd(srow + 2, v.z);
    unsafeAtomicAdd(srow + 3, v.w);
    if (lane == 0) unsafeAtomicAdd(countf + dst, 1.0f);
}

// ---------------------------------------------------------------------------
// Kernel 3: out = (summed/max(count,1)) @ W_l + b_l + x @ W_r
// Fused K=256 WMMA GEMM in fp32 using V_WMMA_F32_16X16X4_F32.
// Block = 256 threads = 8 waves; wave w owns output cols [16w,16w+16) of a
// 16-row tile; blockIdx.x selects the row tile.
// ---------------------------------------------------------------------------
__global__ void sage_gemm_kernel(const float* __restrict__ summed,
                                 const float* __restrict__ countf,
                                 const float* __restrict__ x,
                                 const float* __restrict__ Wl,
                                 const float* __restrict__ bl,
                                 const float* __restrict__ Wr,
                                 float* __restrict__ out,
                                 int nNodes) {
    const int lane   = threadIdx.x & 31;
    const int wave   = threadIdx.x >> 5;
    const int laneLo = lane & 15;
    const int laneHi = lane >> 4;          // 0: K=0,1   1: K=2,3
    const int n0     = wave * 16;

    int row  = blockIdx.x * 16 + laneLo;
    int rowC = row < nNodes ? row : (nNodes - 1);   // clamp loads, keep EXEC full

    const float inv = 1.0f / fmaxf(countf[rowC], 1.0f);
    const float* aRow = summed + (size_t)rowC * DIM + laneHi * 2;
    const float* xRow = x      + (size_t)rowC * DIM + laneHi * 2;

    v8f acc = {};

    // Phase 1: mean-aggregated neighbors through W_l (mean fused into A load)
#pragma unroll
    for (int k0 = 0; k0 < DIM; k0 += 4) {
        v2f a; a.x = aRow[k0] * inv; a.y = aRow[k0 + 1] * inv;
        const int kb = k0 + laneHi * 2;
        v2f b; b.x = Wl[kb * DIM + n0 + laneLo];
               b.y = Wl[(kb + 1) * DIM + n0 + laneLo];
        acc = __builtin_amdgcn_wmma_f32_16x16x4_f32(
            false, a, false, b, (short)0, acc, false, false);
    }

    // Phase 2: root features through W_r
#pragma unroll
    for (int k0 = 0; k0 < DIM; k0 += 4) {
        v2f a; a.x = xRow[k0]; a.y = xRow[k0 + 1];
        const int kb = k0 + laneHi * 2;
        v2f b; b.x = Wr[kb * DIM + n0 + laneLo];
               b.y = Wr[(kb + 1) * DIM + n0 + laneLo];
        acc = __builtin_amdgcn_wmma_f32_16x16x4_f32(
            false, a, false, b, (short)0, acc, false, false);
    }

    const float bias = bl[n0 + laneLo];
    // C/D layout: VGPR r -> row tileM + r + 8*laneHi, col n0 + laneLo
#pragma unroll
    for (int r = 0; r < 8; ++r) {
        int orow = blockIdx.x * 16 + r + laneHi * 8;
        if (orow < nNodes)
            out[(size_t)orow * DIM + n0 + laneLo] = acc[r] + bias;
    }
}

// ---------------------------------------------------------------------------
extern "C" void kernel_launch(void* const* d_in, const int* in_sizes, int n_in,
                              void* d_out, int out_size, void* d_ws, size_t ws_size,
                              hipStream_t stream) {
    const float*     x  = (const float*)d_in[0];
    const long long* ei = (const long long*)d_in[1];   // int64 edge_index [2,E]
    const float*     Wl = (const float*)d_in[2];
    const float*     bl = (const float*)d_in[3];
    const float*     Wr = (const float*)d_in[4];
    float*           out = (float*)d_out;

    const int nNodes = in_sizes[0] / DIM;
    const int nEdges = in_sizes[1] / 2;

    float* summed = (float*)d_ws;                       // [nNodes, 128]
    float* countf = summed + (size_t)nNodes * DIM;      // [nNodes]

    // 1) zero accumulators
    size_t zn = (size_t)nNodes * DIM + (size_t)nNodes;
    sage_zero_kernel<<<(unsigned)((zn + 255) / 256), 256, 0, stream>>>(summed, zn);

    // 2) edge scatter: 8 edges (waves) per 256-thread block
    unsigned eBlocks = (unsigned)((nEdges + 7) / 8);
    sage_scatter_kernel<<<eBlocks, 256, 0, stream>>>(x, ei, summed, countf, nEdges);

    // 3) fused fp32 WMMA GEMM + bias
    unsigned mTiles = (unsigned)((nNodes + 15) / 16);
    sage_gemm_kernel<<<mTiles, 256, 0, stream>>>(summed, countf, x, Wl, bl, Wr,
                                                 out, nNodes);
}